// GCN_str_4612794876644
// MI455X (gfx1250) — compile-verified
//
#include <hip/hip_runtime.h>

typedef __attribute__((ext_vector_type(2))) float v2f;
typedef __attribute__((ext_vector_type(8))) float v8f;

#define IN_H   18
#define OUT_H  11
#define NCLASS 40
#define EMB_K  51   // NCLASS + OUT_H
#define KPAD   52   // 13 * 4
#define NPAD   48   // 3 * 16

// Workspace float layout (N = #nodes):
//  [0,    N)  : deg -> dinv (in place)
//  [N,  20N)  : t1  scatter target, 19 channels (18 features + ones-channel)
//  [20N,38N)  : agg1 = A(x)          (18 ch)
//  [38N,39N)  : rowsum = A(ones)
//  [39N,57N)  : t2  scatter target   (18 ch)
//  [57N,68N)  : emb_b = relu(h2)     (11 ch)
//  [68N, 68N+198)     : W12 = W1@W2  (18x11)
//  [68N+198, 68N+209) : b12 = b1@W2  (11)

__global__ __launch_bounds__(256) void k_init(float* __restrict__ ws, int N) {
  size_t n = (size_t)N;
  size_t i = (size_t)blockIdx.x * 256 + threadIdx.x;
  if (i < n)            ws[i] = 1.0f;          // deg starts at 1 (self-loop)
  else if (i < 20 * n)  ws[i] = 0.0f;          // t1
  else if (i < 38 * n)  ws[i + 19 * n] = 0.0f; // t2 lives at [39N,57N)
}

// W12[18x11] = W1[18x256] @ W2[256x11];  b12[11] = b1[256] @ W2
__global__ __launch_bounds__(256) void k_w12(const float* __restrict__ W1,
                                             const float* __restrict__ W2,
                                             const float* __restrict__ b1,
                                             float* __restrict__ ws, int N) {
  int t = threadIdx.x;
  float* W12 = ws + (size_t)68 * N;
  float* b12 = W12 + IN_H * OUT_H;
  if (t < IN_H * OUT_H) {
    int i = t / OUT_H, j = t % OUT_H;
    float s = 0.0f;
    for (int k = 0; k < 256; ++k) s += W1[i * 256 + k] * W2[k * OUT_H + j];
    W12[t] = s;
  } else if (t < IN_H * OUT_H + OUT_H) {
    int j = t - IN_H * OUT_H;
    float s = 0.0f;
    for (int k = 0; k < 256; ++k) s += b1[k] * W2[k * OUT_H + j];
    b12[j] = s;
  }
}

__global__ __launch_bounds__(256) void k_deg(const int* __restrict__ ei,
                                             float* __restrict__ ws, int N, int E) {
  int e = blockIdx.x * 256 + threadIdx.x;
  if (e >= E) return;
  int dst = ei[E + e];
  if ((unsigned)dst < (unsigned)N) atomicAdd(&ws[dst], 1.0f);
}

__global__ __launch_bounds__(256) void k_dinv(float* __restrict__ ws, int N) {
  int i = blockIdx.x * 256 + threadIdx.x;
  if (i >= N) return;
  float d = ws[i];
  ws[i] = d > 0.0f ? rsqrtf(d) : 0.0f;
}

// t1[dst][c] += dinv[src]*x[src][c] (c<18), t1[dst][18] += dinv[src]
__global__ __launch_bounds__(256) void k_scatter1(const int* __restrict__ ei,
                                                  const float* __restrict__ x,
                                                  float* __restrict__ ws, int N, int E) {
  int e = blockIdx.x * 256 + threadIdx.x;
  if (e >= E) return;
  int src = ei[e], dst = ei[E + e];
  if ((unsigned)src >= (unsigned)N || (unsigned)dst >= (unsigned)N) return;
  float ns = ws[src];
  const float* xr = x + (size_t)src * IN_H;
  float* t = ws + (size_t)N + (size_t)dst * 19;
#pragma unroll
  for (int c = 0; c < IN_H; ++c) atomicAdd(&t[c], ns * xr[c]);
  atomicAdd(&t[IN_H], ns);
}

// agg1 = dinv*(t1 + dinv*x); rowsum = dinv*(t1[18] + dinv)
__global__ __launch_bounds__(256) void k_fin1(const float* __restrict__ x,
                                              float* __restrict__ ws, int N) {
  int i = blockIdx.x * 256 + threadIdx.x;
  if (i >= N) return;
  size_t n = (size_t)N;
  float d = ws[i];
  const float* t = ws + n + (size_t)i * 19;
  float* a1 = ws + 20 * n + (size_t)i * IN_H;
#pragma unroll
  for (int c = 0; c < IN_H; ++c) a1[c] = d * (t[c] + d * x[(size_t)i * IN_H + c]);
  ws[38 * n + i] = d * (t[IN_H] + d);
}

__global__ __launch_bounds__(256) void k_scatter2(const int* __restrict__ ei,
                                                  float* __restrict__ ws, int N, int E) {
  int e = blockIdx.x * 256 + threadIdx.x;
  if (e >= E) return;
  int src = ei[e], dst = ei[E + e];
  if ((unsigned)src >= (unsigned)N || (unsigned)dst >= (unsigned)N) return;
  size_t n = (size_t)N;
  float ns = ws[src];
  const float* a1 = ws + 20 * n + (size_t)src * IN_H;
  float* t = ws + 39 * n + (size_t)dst * IN_H;
#pragma unroll
  for (int c = 0; c < IN_H; ++c) atomicAdd(&t[c], ns * a1[c]);
}

// agg2 = dinv*(t2 + dinv*agg1); emb_b = relu(agg2@W12 + rowsum*b12 + b2)
__global__ __launch_bounds__(256) void k_fin2(const float* __restrict__ b2,
                                              float* __restrict__ ws, int N) {
  int i = blockIdx.x * 256 + threadIdx.x;
  if (i >= N) return;
  size_t n = (size_t)N;
  float d = ws[i];
  const float* a1 = ws + 20 * n + (size_t)i * IN_H;
  const float* t2 = ws + 39 * n + (size_t)i * IN_H;
  const float* W12 = ws + 68 * n;
  const float* b12 = W12 + IN_H * OUT_H;
  float rs = ws[38 * n + i];
  float agg2[IN_H];
#pragma unroll
  for (int c = 0; c < IN_H; ++c) agg2[c] = d * (t2[c] + d * a1[c]);
  float* eb = ws + 57 * n + (size_t)i * OUT_H;
#pragma unroll
  for (int j = 0; j < OUT_H; ++j) {
    float y = rs * b12[j] + b2[j];
#pragma unroll
    for (int c = 0; c < IN_H; ++c) y += agg2[c] * W12[c * OUT_H + j];
    eb[j] = fmaxf(y, 0.0f);
  }
}

// out[N x 40] = [emb_a | emb_b] @ Wc + bc, via V_WMMA_F32_16X16X4_F32.
// One wave = 16 nodes; 3 N-tiles (48 cols) x 13 K-steps (52). Wc zero-padded in LDS.
__global__ __launch_bounds__(256) void k_final(const float* __restrict__ emb_a,
                                               const float* __restrict__ emb_b,
                                               const float* __restrict__ Wc,
                                               const float* __restrict__ bc,
                                               float* __restrict__ out, int N) {
  __shared__ float sW[KPAD * NPAD];
  __shared__ float sB[NPAD];
  int tid = threadIdx.x;
  for (int idx = tid; idx < KPAD * NPAD; idx += 256) {
    int kk = idx / NPAD, nn = idx % NPAD;
    sW[idx] = (kk < EMB_K && nn < NCLASS) ? Wc[kk * NCLASS + nn] : 0.0f;
  }
  if (tid < NPAD) sB[tid] = (tid < NCLASS) ? bc[tid] : 0.0f;
  __syncthreads();

  int wave = tid >> 5, lane = tid & 31;
  int row = lane & 15, sel = lane >> 4;           // A: lanes 0-15 hold K pair {0,1}, 16-31 hold {2,3}
  int node0 = (blockIdx.x * 8 + wave) * 16;
  int node = node0 + row;
  bool ok = node < N;
  int nc = ok ? node : 0;

  v8f acc0 = {}, acc1 = {}, acc2 = {};
  for (int ks = 0; ks < 13; ++ks) {
    int kb = ks * 4 + sel * 2;
    int k1 = kb + 1;                               // kb<=50, k1<=51 (row 51 of sW is zero pad)
    float a0 = (kb < NCLASS) ? emb_a[(size_t)nc * NCLASS + kb]
             : (kb < EMB_K  ? emb_b[(size_t)nc * OUT_H + (kb - NCLASS)] : 0.0f);
    float a1 = (k1 < NCLASS) ? emb_a[(size_t)nc * NCLASS + k1]
             : (k1 < EMB_K  ? emb_b[(size_t)nc * OUT_H + (k1 - NCLASS)] : 0.0f);
    v2f a; a.x = ok ? a0 : 0.0f; a.y = ok ? a1 : 0.0f;

    v2f b0, b1v, b2v;
    b0.x  = sW[kb * NPAD +  0 + row];  b0.y  = sW[k1 * NPAD +  0 + row];
    b1v.x = sW[kb * NPAD + 16 + row];  b1v.y = sW[k1 * NPAD + 16 + row];
    b2v.x = sW[kb * NPAD + 32 + row];  b2v.y = sW[k1 * NPAD + 32 + row];

    acc0 = __builtin_amdgcn_wmma_f32_16x16x4_f32(false, a, false, b0,  (short)0, acc0, false, false);
    acc1 = __builtin_amdgcn_wmma_f32_16x16x4_f32(false, a, false, b1v, (short)0, acc1, false, false);
    acc2 = __builtin_amdgcn_wmma_f32_16x16x4_f32(false, a, false, b2v, (short)0, acc2, false, false);
  }

  // C/D layout: vgpr v holds M = v + sel*8, col N = row (+ 16*tile)
#pragma unroll
  for (int v = 0; v < 8; ++v) {
    int m = v + sel * 8;
    int nd = node0 + m;
    if (nd < N) {
      int n0 = row;
      if (n0 < NCLASS)      out[(size_t)nd * NCLASS + n0]      = acc0[v] + sB[n0];
      int n1 = 16 + row;
      if (n1 < NCLASS)      out[(size_t)nd * NCLASS + n1]      = acc1[v] + sB[n1];
      int n2 = 32 + row;
      if (n2 < NCLASS)      out[(size_t)nd * NCLASS + n2]      = acc2[v] + sB[n2];
    }
  }
}

extern "C" void kernel_launch(void* const* d_in, const int* in_sizes, int n_in,
                              void* d_out, int out_size, void* d_ws, size_t ws_size,
                              hipStream_t stream) {
  const float* stc_enc = (const float*)d_in[0];
  const int*   ei      = (const int*)d_in[1];   // [2,E] int
  const float* emb_a   = (const float*)d_in[2];
  const float* W1      = (const float*)d_in[3];
  const float* b1      = (const float*)d_in[4];
  const float* W2      = (const float*)d_in[5];
  const float* b2      = (const float*)d_in[6];
  const float* Wc      = (const float*)d_in[7];
  const float* bc      = (const float*)d_in[8];
  float* out = (float*)d_out;
  float* ws  = (float*)d_ws;

  const int N = in_sizes[0] / IN_H;
  const int E = in_sizes[1] / 2;

  size_t initElems = (size_t)38 * N;
  int gInit = (int)((initElems + 255) / 256);
  int gN    = (N + 255) / 256;
  int gE    = (E + 255) / 256;
  int gF    = (N + 127) / 128;   // 8 waves x 16 nodes per block

  k_init    <<<gInit, 256, 0, stream>>>(ws, N);
  k_w12     <<<1,     256, 0, stream>>>(W1, W2, b1, ws, N);
  k_deg     <<<gE,    256, 0, stream>>>(ei, ws, N, E);
  k_dinv    <<<gN,    256, 0, stream>>>(ws, N);
  k_scatter1<<<gE,    256, 0, stream>>>(ei, stc_enc, ws, N, E);
  k_fin1    <<<gN,    256, 0, stream>>>(stc_enc, ws, N);
  k_scatter2<<<gE,    256, 0, stream>>>(ei, ws, N, E);
  k_fin2    <<<gN,    256, 0, stream>>>(b2, ws, N);
  k_final   <<<gF,    256, 0, stream>>>(emb_a, ws + (size_t)57 * N, Wc, bc, out, N);
}